// Model_77360950935954
// MI455X (gfx1250) — compile-verified
//
#include <hip/hip_runtime.h>
#include <stdint.h>

// Radon transform for MI455X (gfx1250):
//  - one workgroup per (batch, 8-angle group)
//  - whole 256x256 f32 image staged into LDS via TDM tensor_load_to_lds with
//    row padding (stride 257 dwords) to avoid LDS bank conflicts
//  - ray integration: bilinear 4-tap gather from LDS, h-reduction done with
//    V_WMMA_F32_16X16X4_F32 using an all-ones A matrix (column-sum trick),
//    full fp32 precision matching the reference.

#define NPIX       256
#define ROWSTRIDE  257          // +1 dword pad: bank = (y*257+x)%64 spreads over y
#define NANG_PER_WG 8
#define THREADS    512          // 16 wave32 -> 16 detector tiles of 16

typedef __attribute__((ext_vector_type(2))) float    v2f;
typedef __attribute__((ext_vector_type(8))) float    v8f;
typedef __attribute__((ext_vector_type(4))) unsigned v4u;
typedef __attribute__((ext_vector_type(4))) int      v4i;
typedef __attribute__((ext_vector_type(8))) int      v8i;

static __device__ __forceinline__ int iclamp(int v, int lo, int hi) {
    return v < lo ? lo : (v > hi ? hi : v);
}

// Bilinear sample with zero padding outside [0,255]^2, from LDS image
// stored with row stride ROWSTRIDE dwords.
static __device__ __forceinline__ float bilin(const float* __restrict__ s,
                                              float px, float py) {
    const float x0f = floorf(px), y0f = floorf(py);
    const float wx1 = px - x0f,   wy1 = py - y0f;
    const float wx0 = 1.0f - wx1, wy0 = 1.0f - wy1;
    // per-corner validity (x1 = x0+1, y1 = y0+1)
    const float vx0 = (x0f >= 0.0f  && x0f <= 255.0f) ? 1.0f : 0.0f;
    const float vx1 = (x0f >= -1.0f && x0f <= 254.0f) ? 1.0f : 0.0f;
    const float vy0 = (y0f >= 0.0f  && y0f <= 255.0f) ? 1.0f : 0.0f;
    const float vy1 = (y0f >= -1.0f && y0f <= 254.0f) ? 1.0f : 0.0f;
    const int x0 = (int)x0f, y0 = (int)y0f;
    const int x0c = iclamp(x0, 0, 255),     x1c = iclamp(x0 + 1, 0, 255);
    const int y0c = iclamp(y0, 0, 255),     y1c = iclamp(y0 + 1, 0, 255);
    const float v00 = s[y0c * ROWSTRIDE + x0c];
    const float v01 = s[y0c * ROWSTRIDE + x1c];
    const float v10 = s[y1c * ROWSTRIDE + x0c];
    const float v11 = s[y1c * ROWSTRIDE + x1c];
    return (wy0 * vy0) * ((wx0 * vx0) * v00 + (wx1 * vx1) * v01)
         + (wy1 * vy1) * ((wx0 * vx0) * v10 + (wx1 * vx1) * v11);
}

extern "C" __global__ __launch_bounds__(THREADS, 1)
void radon_wmma_kernel(const float* __restrict__ x,
                       const float* __restrict__ angles,
                       float* __restrict__ out) {
    extern __shared__ float smem[];          // 256 rows * 257 dwords
    const int tid  = threadIdx.x;
    const int lane = tid & 31;
    const int wave = tid >> 5;
    const int b    = blockIdx.x >> 5;        // batch 0..31
    const int g    = blockIdx.x & 31;        // angle group 0..31

    // ---------------- stage image b into LDS ----------------
#if __has_builtin(__builtin_amdgcn_tensor_load_to_lds)
    if (wave == 0) {
        const uint64_t ga = (uint64_t)(uintptr_t)(x + (size_t)b * NPIX * NPIX);
        const unsigned lds_base = (unsigned)(uintptr_t)(void*)&smem[0];
        v4u g0;
        g0[0] = 1u;                                    // count=1, user descriptor
        g0[1] = lds_base;                              // lds_addr (bytes)
        g0[2] = (unsigned)ga;                          // global_addr[31:0]
        g0[3] = (unsigned)(ga >> 32) | (2u << 30);     // global_addr[56:32] | type=2
        v8i g1;
        // data_size=2 (4B), pad_enable=1, pad_interval=7 (256 dw), pad_amount=0 (1 dw)
        g1[0] = (int)((2u << 16) | (1u << 20) | (7u << 22));
        g1[1] = (int)(256u << 16);   // [63:48]  tensor_dim0   = 256
        g1[2] = (int)(256u << 16);   // [95:80]  tensor_dim1   = 256
        g1[3] = (int)(256u << 16);   // [127:112] tile_dim0    = 256
        g1[4] = (int)(256u);         // [143:128] tile_dim1    = 256, tile_dim2 = 0
        g1[5] = (int)(256u);         // tensor_dim0_stride[31:0] = 256
        g1[6] = 0;                   // stride hi / dim1_stride lo
        g1[7] = 0;
        const v4i gz = {0, 0, 0, 0};
#if defined(__clang_major__) && (__clang_major__ >= 23)
        const v8i gz8 = {0, 0, 0, 0, 0, 0, 0, 0};
        __builtin_amdgcn_tensor_load_to_lds(g0, g1, gz, gz, gz8, 0);
#else
        __builtin_amdgcn_tensor_load_to_lds(g0, g1, gz, gz, 0);
#endif
#if __has_builtin(__builtin_amdgcn_s_wait_tensorcnt)
        __builtin_amdgcn_s_wait_tensorcnt(0);
#endif
    }
#else
    {   // manual staging fallback: float4 loads -> LDS with stride 257
        const float4* src = (const float4*)(x + (size_t)b * NPIX * NPIX);
        for (int i = tid; i < NPIX * (NPIX / 4); i += THREADS) {
            const int row = i >> 6;
            const int c4  = (i & 63) << 2;
            const float4 v = src[i];
            float* d = &smem[row * ROWSTRIDE + c4];
            d[0] = v.x; d[1] = v.y; d[2] = v.z; d[3] = v.w;
        }
    }
#endif
    __syncthreads();

    // ---------------- ray integration ----------------
    // wave -> 16-detector tile; lane n (n&15) -> detector column; half-wave
    // selects h-phase {0,1} vs {2,3} inside each 4-step WMMA K-chunk.
    const int   wtile = wave;
    const float wf    = (float)(wtile * 16 + (lane & 15));
    const float o0    = (lane < 16) ? 0.0f : 2.0f;
    const v2f   ones  = {1.0f, 1.0f};

    for (int i = 0; i < NANG_PER_WG; ++i) {
        const int a   = g * NANG_PER_WG + i;
        const float t = angles[a];
        const float ct = cosf(t);
        const float st = sinf(t);
        // px(h) = pxw - st*h ; py(h) = pyw + ct*h
        const float pxw = 127.5f + ct * (wf - 127.5f) + 127.5f * st;
        const float pyw = 127.5f + st * (wf - 127.5f) - 127.5f * ct;

        v8f acc = {0.f, 0.f, 0.f, 0.f, 0.f, 0.f, 0.f, 0.f};
#if !__has_builtin(__builtin_amdgcn_wmma_f32_16x16x4_f32)
        float acc0 = 0.0f;
#endif
        for (int hc = 0; hc < 64; ++hc) {
            const float h0 = (float)(4 * hc) + o0;
            const float h1 = h0 + 1.0f;
            const float sA = bilin(smem, fmaf(-st, h0, pxw), fmaf(ct, h0, pyw));
            const float sB = bilin(smem, fmaf(-st, h1, pxw), fmaf(ct, h1, pyw));
#if __has_builtin(__builtin_amdgcn_wmma_f32_16x16x4_f32)
            v2f bm; bm[0] = sA; bm[1] = sB;
            // A = ones => D[m,n] = sum_k B[k,n] + C[m,n]  (column-sum reduction)
            acc = __builtin_amdgcn_wmma_f32_16x16x4_f32(
                false, ones, false, bm, (short)0, acc, false, false);
#else
            acc0 += sA + sB;
#endif
        }
#if __has_builtin(__builtin_amdgcn_wmma_f32_16x16x4_f32)
        const float res = acc[0];     // D row 0: lanes 0..15 hold N=0..15
#else
        const float res = acc0 + __shfl_down(acc0, 16, 32);
#endif
        if (lane < 16) {
            out[((size_t)b * 256 + a) * 256 + (size_t)(wtile * 16 + lane)] = res;
        }
    }
}

extern "C" void kernel_launch(void* const* d_in, const int* in_sizes, int n_in,
                              void* d_out, int out_size, void* d_ws, size_t ws_size,
                              hipStream_t stream) {
    (void)in_sizes; (void)n_in; (void)d_ws; (void)ws_size; (void)out_size;
    const float* x      = (const float*)d_in[0];
    const float* angles = (const float*)d_in[1];
    float*       out    = (float*)d_out;

    dim3 grid(32 * 32);      // batch x angle-group
    dim3 block(THREADS);
    const size_t shmem = (size_t)NPIX * ROWSTRIDE * sizeof(float);  // 257 KB
    hipLaunchKernelGGL(radon_wmma_kernel, grid, block, shmem, stream,
                       x, angles, out);
}